// Match_21466246545847
// MI455X (gfx1250) — compile-verified
//
#include <hip/hip_runtime.h>
#include <hip/hip_bf16.h>

typedef __attribute__((ext_vector_type(2))) float v2f;
typedef __attribute__((ext_vector_type(8))) float v8f;

#define ZNEAR 0.1f
#define ZFAR  100.0f

// One lane per point. Each wave of 32 lanes projects 32 points of one batch via
// two V_WMMA_F32_16X16X4_F32 ops (pose in A rows 0-3 -> results in lanes 0-15;
// pose in A rows 8-11 -> results in lanes 16-31), then does the bilinear beta
// sample and the packed-u64 atomicMax z-buffer scatter.
__global__ void proj_scatter_kernel(
    const float* __restrict__ beta,   // [b,1,h,w]
    const float* __restrict__ pp,     // [b,7,n]
    const float* __restrict__ conf,   // [b,n]
    const float* __restrict__ pose,   // [b,4,4]
    const float* __restrict__ K,      // [b,3,3]
    const int* __restrict__ hp, const int* __restrict__ wp,
    float* __restrict__ out_conf,     // [b,1,n]
    unsigned long long* __restrict__ zbuf) // [b, 2h, 2w] packed (conf_bits<<32)|idx
{
    const int h = *hp, w = *wp;
    const int n = h * w;
    const int W2 = w * 2, H2 = h * 2;

    const int gid  = blockIdx.x * blockDim.x + threadIdx.x;
    const int b    = gid / n;
    const int i    = gid - b * n;          // point index within batch
    const int lane = threadIdx.x & 31;
    const int base = i - lane;             // 32-aligned chunk base (n % 256 == 0)

    const float* Pb    = pp   + (size_t)b * 7 * n;
    const float* poseb = pose + b * 16;
    const int m  = lane & 15;              // matrix row / B column within half
    const int kh = lane >> 4;              // 0: holds K=0,1 ; 1: holds K=2,3

    // A1: pose in rows 0-3 (for points base..base+15)
    float a1x = (m < 4) ? poseb[m * 4 + (kh ? 2 : 0)] : 0.0f;
    float a1y = (m < 4) ? poseb[m * 4 + (kh ? 3 : 1)] : 0.0f;
    // A2: pose in rows 8-11 (for points base+16..base+31 -> lanes 16-31 of D)
    float a2x = (m >= 8 && m < 12) ? poseb[(m - 8) * 4 + (kh ? 2 : 0)] : 0.0f;
    float a2y = (m >= 8 && m < 12) ? poseb[(m - 8) * 4 + (kh ? 3 : 1)] : 0.0f;

    // B columns: lanes 0-15 supply K=0,1 (X,Y); lanes 16-31 supply K=2,3 (Z,W)
    const size_t pc1i = (size_t)(base + m);
    const size_t pc2i = pc1i + 16;
    float b1x = kh ? Pb[2 * (size_t)n + pc1i] : Pb[0 * (size_t)n + pc1i];
    float b1y = kh ? Pb[3 * (size_t)n + pc1i] : Pb[1 * (size_t)n + pc1i];
    float b2x = kh ? Pb[2 * (size_t)n + pc2i] : Pb[0 * (size_t)n + pc2i];
    float b2y = kh ? Pb[3 * (size_t)n + pc2i] : Pb[1 * (size_t)n + pc2i];

    v2f A1 = {a1x, a1y}, A2 = {a2x, a2y};
    v2f B1 = {b1x, b1y}, B2 = {b2x, b2y};
    v8f Cz = {0.f, 0.f, 0.f, 0.f, 0.f, 0.f, 0.f, 0.f};

    v8f D1 = __builtin_amdgcn_wmma_f32_16x16x4_f32(false, A1, false, B1,
                                                   (short)0, Cz, false, false);
    v8f D2 = __builtin_amdgcn_wmma_f32_16x16x4_f32(false, A2, false, B2,
                                                   (short)0, Cz, false, false);

    // lane < 16: D1 rows 0..2 at N=lane ; lane >= 16: D2 rows 8..10 at N=lane-16
    float pc0 = kh ? D2[0] : D1[0];
    float pc1 = kh ? D2[1] : D1[1];
    float pc2 = kh ? D2[2] : D1[2];

    const float* Kb = K + b * 9;
    const float fx = Kb[0], cx = Kb[2], fy = Kb[4], cy = Kb[5];
    const float z   = fabsf(pc2);
    const float inv = 1.0f / z;
    const float xc  = pc0 * fx * inv + cx;
    const float yc  = pc1 * fy * inv + cy;

    // Bilinear sample of beta at (ix=xc, iy=(h-1)-yc), zeros padding.
    const float* Bb = beta + (size_t)b * n;
    const float ix = xc;
    const float iy = (float)(h - 1) - yc;
    const float x0f = floorf(ix), y0f = floorf(iy);
    const float wx = ix - x0f, wy = iy - y0f;
    const int x0 = (int)x0f, y0 = (int)y0f;
    const int x1 = x0 + 1,   y1 = y0 + 1;
    const int x0c = min(max(x0, 0), w - 1), x1c = min(max(x1, 0), w - 1);
    const int y0c = min(max(y0, 0), h - 1), y1c = min(max(y1, 0), h - 1);
    const bool bx0 = (x0 >= 0) & (x0 < w), bx1 = (x1 >= 0) & (x1 < w);
    const bool by0 = (y0 >= 0) & (y0 < h), by1 = (y1 >= 0) & (y1 < h);
    const float v00 = (bx0 & by0) ? Bb[(size_t)y0c * w + x0c] : 0.0f;
    const float v10 = (bx1 & by0) ? Bb[(size_t)y0c * w + x1c] : 0.0f;
    const float v01 = (bx0 & by1) ? Bb[(size_t)y1c * w + x0c] : 0.0f;
    const float v11 = (bx1 & by1) ? Bb[(size_t)y1c * w + x1c] : 0.0f;
    const float cs = v00 * (1.0f - wx) * (1.0f - wy) + v10 * wx * (1.0f - wy)
                   + v01 * (1.0f - wx) * wy          + v11 * wx * wy;
    out_conf[(size_t)b * n + i] = cs;

    // Scatter with round-half-to-even (matches jnp.round via v_rndne).
    const float xsf = __builtin_rintf(xc * 2.0f);
    const float ysf = __builtin_rintf(yc * 2.0f);
    const float c = conf[(size_t)b * n + i];
    const bool ok = (xsf >= 0.0f) && (xsf <= (float)(W2 - 1))
                 && (ysf >= 0.0f) && (ysf <= (float)(H2 - 1))
                 && (z >= ZNEAR) && (z <= ZFAR) && (c > 0.0f);
    if (ok) {
        const int xs = (int)xsf, ys = (int)ysf;
        const unsigned long long pk =
            ((unsigned long long)__float_as_uint(c) << 32) | (unsigned int)i;
        atomicMax(zbuf + (size_t)b * H2 * W2 + (size_t)ys * W2 + xs, pk);
    }
}

// One thread per output cell [b,y,x]: read its 2x2 hi-res window (pre-flip row
// yr = h-1-y), alpha mask collapses to alpha[b,0,y,x], argmax (ties -> first).
__global__ void pool_kernel(
    const float* __restrict__ alpha,  // [b,1,h,w]
    const int* __restrict__ hp, const int* __restrict__ wp,
    const unsigned long long* __restrict__ zbuf,
    float* __restrict__ out_match, float* __restrict__ out_idx)
{
    const int h = *hp, w = *wp;
    const int hw = h * w;
    const int W2 = 2 * w;

    const int gid = blockIdx.x * blockDim.x + threadIdx.x;
    const int b = gid / hw;
    const int r = gid - b * hw;
    const int y = r / w;
    const int x = r - y * w;
    const int yr = h - 1 - y;

    const unsigned long long* zb = zbuf + (size_t)b * 4 * (size_t)hw;
    const size_t o = (size_t)(2 * yr) * W2 + 2 * x;
    unsigned long long pv[4];
    pv[0] = zb[o];      pv[1] = zb[o + 1];
    pv[2] = zb[o + W2]; pv[3] = zb[o + W2 + 1];

    const bool ak = alpha[(size_t)b * hw + r] > 0.0f;

    float best = 0.0f;
    int besta = 0;
    bool any = false;
#pragma unroll
    for (int a = 0; a < 4; ++a) {
        const bool keep = ak && (pv[a] != 0ULL);
        const float cm = keep ? __uint_as_float((unsigned)(pv[a] >> 32)) : 0.0f;
        any = any || keep;
        if (cm > best) { best = cm; besta = a; }
    }
    const bool keepw = ak && (pv[besta] != 0ULL);
    const int idx = keepw ? (int)(unsigned)pv[besta] : -1;

    out_match[gid] = any ? 1.0f : 0.0f;
    out_idx[gid]   = (float)idx;   // int region, value-cast into concat buffer
}

extern "C" void kernel_launch(void* const* d_in, const int* in_sizes, int n_in,
                              void* d_out, int out_size, void* d_ws, size_t ws_size,
                              hipStream_t stream) {
    const float* alpha = (const float*)d_in[0];
    const float* beta  = (const float*)d_in[1];
    const float* pp    = (const float*)d_in[2];
    const float* conf  = (const float*)d_in[3];
    const float* pose  = (const float*)d_in[4];
    const float* K     = (const float*)d_in[5];
    const int*   hp    = (const int*)d_in[6];
    const int*   wp    = (const int*)d_in[7];

    const int BHW = in_sizes[0];  // b*h*w  (alpha)
    const int BN  = in_sizes[3];  // b*n == b*h*w (conf)

    unsigned long long* zbuf = (unsigned long long*)d_ws;
    const size_t zbytes = (size_t)BHW * 4 * sizeof(unsigned long long); // b*2h*2w u64
    hipMemsetAsync(d_ws, 0, zbytes, stream);

    float* out_match = (float*)d_out;
    float* out_idx   = out_match + BHW;
    float* out_conf  = out_idx + BHW;

    const int threads = 256;
    proj_scatter_kernel<<<BN / threads, threads, 0, stream>>>(
        beta, pp, conf, pose, K, hp, wp, out_conf, zbuf);
    pool_kernel<<<BHW / threads, threads, 0, stream>>>(
        alpha, hp, wp, zbuf, out_match, out_idx);
}